// Sampler_44126493999596
// MI455X (gfx1250) — compile-verified
//
#include <hip/hip_runtime.h>
#include <hip/hip_bf16.h>
#include <stdint.h>

// ---------------------------------------------------------------------------
// Top-p sampling without a sort (one workgroup per row, 8 wave32 each):
//   phase 1: stream logits via 16-deep async global->LDS pipeline, build LDS
//            histogram of w=exp(s) keyed by s = logit/T, and reduce Z.
//   phase 2: parallel top-down suffix scan of histogram -> crossing bin,
//            one refinement pass inside that bin (2048 sub-bins, L2-resident).
//   phase 3: stream noise via 16-deep async pipeline + logits from L2,
//            argmax of exp(s)/max(-log(u),1e-10) over { s >= threshold }.
// ---------------------------------------------------------------------------

#define B_ROWS   256
#define VOCAB    128000
#define THREADS  256
#define NBINS    2048
#define BPT      (NBINS / THREADS)      // bins per thread = 8
#define CHUNK    (THREADS * 4)          // 1024 floats per chunk
#define NCHUNK   (VOCAB / CHUNK)        // 125 chunks, exact
#define NBUF     16                     // async pipeline depth (power of 2)
#define SMIN_F   (-16.0f)
#define SMAX_F   ( 16.0f)

#if defined(__has_builtin)
#if __has_builtin(__builtin_amdgcn_global_load_async_to_lds_b128) && \
    __has_builtin(__builtin_amdgcn_s_wait_asynccnt)
#define USE_ASYNC_LDS 1
#endif
#endif
#ifndef USE_ASYNC_LDS
#define USE_ASYNC_LDS 0
#endif

#if USE_ASYNC_LDS
// Builtin signature (from clang diagnostic): pointers to int4 in
// addrspace(1) (global) and addrspace(3) (LDS).
typedef int v4i_t __attribute__((vector_size(16)));
typedef __attribute__((address_space(1))) v4i_t GV4;
typedef __attribute__((address_space(3))) v4i_t LV4;
#endif

__device__ __forceinline__ unsigned long long pack_key(float score, int idx) {
  // larger score wins; tie -> smaller idx wins (matches jnp.argmax first-hit)
  return ((unsigned long long)__float_as_uint(score) << 32) |
         (unsigned long long)(0xFFFFFFFFu - (unsigned)idx);
}

__global__ __launch_bounds__(THREADS)
void topp_sample_kernel(const float* __restrict__ logits,
                        const float* __restrict__ temps,
                        const float* __restrict__ topps,
                        const float* __restrict__ noise,
                        int* __restrict__ out) {
  const int row = blockIdx.x;
  const int tid = threadIdx.x;

  const float invT = 1.0f / temps[row];
  const float topp = topps[row];
  const float* __restrict__ lrow = logits + (size_t)row * VOCAB;
  const float* __restrict__ nrow = noise  + (size_t)row * VOCAB;

  __shared__ float hist[NBINS];
  __shared__ __align__(16) float stage[NBUF][CHUNK];   // 64 KB circular stage
  __shared__ __align__(16) float scrap[CHUNK];         // dummy async target
  __shared__ float part[THREADS];
  __shared__ unsigned long long keys[THREADS];
  __shared__ int   sh_bin;
  __shared__ float sh_rem;

  const float binW    = (SMAX_F - SMIN_F) / (float)NBINS;
  const float invBinW = 1.0f / binW;

  for (int i = tid; i < NBINS; i += THREADS) hist[i] = 0.0f;
  __syncthreads();

  // ---- block-wide "find crossing bin" given a target mass (top-down) ------
  auto find_cross = [&](float target) {
    __syncthreads();                       // hist stable, part free
    float local = 0.0f;
#pragma unroll
    for (int k = 0; k < BPT; ++k) local += hist[tid * BPT + k];
    part[tid] = local;
    __syncthreads();
    // inclusive suffix scan over per-thread partials (Hillis-Steele)
    for (int off = 1; off < THREADS; off <<= 1) {
      float v = part[tid];
      if (tid + off < THREADS) v += part[tid + off];
      __syncthreads();
      part[tid] = v;
      __syncthreads();
    }
    if (tid == 0) { sh_bin = -1; sh_rem = 0.0f; }
    __syncthreads();
    float above = (tid == THREADS - 1) ? 0.0f : part[tid + 1];
#pragma unroll
    for (int k = BPT - 1; k >= 0; --k) {
      const int j = tid * BPT + k;
      const float h = hist[j];
      if (above <= target && above + h > target) {
        sh_bin = j;                 // at most one thread/bin satisfies this
        sh_rem = target - above;
      }
      above += h;
    }
    __syncthreads();
  };

  // ---- phase 1: histogram + Z (16-deep async global->LDS pipeline) --------
  float zsum = 0.0f;
  auto accum = [&](float x) {
    const float s  = x * invT;
    const float sw = fminf(fmaxf(s, -30.0f), 30.0f);
    const float w  = __expf(sw);
    zsum += w;
    int bin = (int)((s - SMIN_F) * invBinW);
    bin = min(max(bin, 0), NBINS - 1);
    atomicAdd(&hist[bin], w);              // ds_add_f32
  };

#if USE_ASYNC_LDS
#pragma unroll
  for (int p = 0; p < NBUF - 1; ++p) {     // prologue: 15 chunks in flight
    __builtin_amdgcn_global_load_async_to_lds_b128(
        (GV4*)(lrow + p * CHUNK + tid * 4), (LV4*)&stage[p][tid * 4], 0, 0);
  }
  for (int c = 0; c < NCHUNK; ++c) {
    const int nx = c + NBUF - 1;
    if (nx < NCHUNK) {
      __builtin_amdgcn_global_load_async_to_lds_b128(
          (GV4*)(lrow + nx * CHUNK + tid * 4),
          (LV4*)&stage[nx & (NBUF - 1)][tid * 4], 0, 0);
    } else {
      // keep the in-order ASYNCcnt ledger running; never aliases live stage
      __builtin_amdgcn_global_load_async_to_lds_b128(
          (GV4*)(lrow + tid * 4), (LV4*)&scrap[tid * 4], 0, 0);
    }
    __builtin_amdgcn_s_wait_asynccnt(NBUF - 1);   // chunk c landed
    __asm__ volatile("" ::: "memory");
    const float4 v = *(const float4*)&stage[c & (NBUF - 1)][tid * 4];
    accum(v.x); accum(v.y); accum(v.z); accum(v.w);
  }
#else
  for (int c = 0; c < NCHUNK; ++c) {
    const float4 v = *(const float4*)(lrow + c * CHUNK + tid * 4);
    accum(v.x); accum(v.y); accum(v.z); accum(v.w);
  }
#endif

  // reduce Z across block
  __syncthreads();
  part[tid] = zsum;
  __syncthreads();
  for (int off = THREADS / 2; off > 0; off >>= 1) {
    if (tid < off) part[tid] += part[tid + off];
    __syncthreads();
  }
  const float Z = part[0];
  const float target = topp * Z;

  // ---- phase 2: coarse crossing bin, then one refinement level ------------
  find_cross(target);
  const int   bin1 = sh_bin;
  const float rem1 = sh_rem;

  float thresh;
  if (bin1 < 0) {
    thresh = -1e30f;                       // keep everything (shouldn't hit)
  } else {
    const float lo1    = SMIN_F + bin1 * binW;
    const float hi1    = lo1 + binW;
    const float subW   = binW / (float)NBINS;
    const float invSub = 1.0f / subW;

    __syncthreads();
    for (int i = tid; i < NBINS; i += THREADS) hist[i] = 0.0f;
    __syncthreads();

    for (int c = 0; c < NCHUNK; ++c) {     // L2-resident re-read
      const int base = c * CHUNK + tid * 4;
      __builtin_prefetch(lrow + base + 4 * CHUNK, 0, 1);   // global_prefetch_b8
      const float4 v = *(const float4*)(lrow + base);
      const float xs[4] = {v.x, v.y, v.z, v.w};
#pragma unroll
      for (int k = 0; k < 4; ++k) {
        const float s = xs[k] * invT;
        if (s >= lo1 && s < hi1) {
          int sb = (int)((s - lo1) * invSub);
          sb = min(max(sb, 0), NBINS - 1);
          atomicAdd(&hist[sb], __expf(s));
        }
      }
    }
    find_cross(rem1);
    const int bin2 = sh_bin;
    thresh = (bin2 < 0) ? lo1 : lo1 + bin2 * subW;
  }

  // ---- phase 3: argmax of w/e over kept set -------------------------------
  // noise is the cold HBM stream -> 16-deep async pipeline; logits hit L2.
  unsigned long long best = 0ull;
  auto score_at = [&](float x, float u, int idx) {
    const float s = x * invT;
    if (s >= thresh) {
      const float sw = fminf(fmaxf(s, -30.0f), 30.0f);
      const float w  = __expf(sw);
      const float e  = fmaxf(-__logf(u), 1e-10f);
      const unsigned long long key = pack_key(w / e, idx);
      best = (key > best) ? key : best;
    }
  };

#if USE_ASYNC_LDS
#pragma unroll
  for (int p = 0; p < NBUF - 1; ++p) {
    __builtin_amdgcn_global_load_async_to_lds_b128(
        (GV4*)(nrow + p * CHUNK + tid * 4), (LV4*)&stage[p][tid * 4], 0, 0);
  }
  for (int c = 0; c < NCHUNK; ++c) {
    const int nx = c + NBUF - 1;
    if (nx < NCHUNK) {
      __builtin_amdgcn_global_load_async_to_lds_b128(
          (GV4*)(nrow + nx * CHUNK + tid * 4),
          (LV4*)&stage[nx & (NBUF - 1)][tid * 4], 0, 0);
    } else {
      __builtin_amdgcn_global_load_async_to_lds_b128(
          (GV4*)(nrow + tid * 4), (LV4*)&scrap[tid * 4], 0, 0);
    }
    const int base = c * CHUNK + tid * 4;
    __builtin_prefetch(lrow + base + 2 * CHUNK, 0, 1);
    const float4 lv = *(const float4*)(lrow + base);       // L2 hit
    __builtin_amdgcn_s_wait_asynccnt(NBUF - 1);            // noise chunk c in
    __asm__ volatile("" ::: "memory");
    const float4 uv = *(const float4*)&stage[c & (NBUF - 1)][tid * 4];
    score_at(lv.x, uv.x, base + 0);
    score_at(lv.y, uv.y, base + 1);
    score_at(lv.z, uv.z, base + 2);
    score_at(lv.w, uv.w, base + 3);
  }
#else
  for (int c = 0; c < NCHUNK; ++c) {
    const int base = c * CHUNK + tid * 4;
    const float4 lv = *(const float4*)(lrow + base);
    const float4 uv = *(const float4*)(nrow + base);
    score_at(lv.x, uv.x, base + 0);
    score_at(lv.y, uv.y, base + 1);
    score_at(lv.z, uv.z, base + 2);
    score_at(lv.w, uv.w, base + 3);
  }
#endif

  keys[tid] = best;
  __syncthreads();
  for (int off = THREADS / 2; off > 0; off >>= 1) {
    if (tid < off) {
      const unsigned long long o = keys[tid + off];
      if (o > keys[tid]) keys[tid] = o;
    }
    __syncthreads();
  }
  if (tid == 0) {
    const unsigned long long k = keys[0];
    out[row] = (int)(0xFFFFFFFFu - (unsigned)(k & 0xFFFFFFFFull));
  }
}

extern "C" void kernel_launch(void* const* d_in, const int* in_sizes, int n_in,
                              void* d_out, int out_size, void* d_ws, size_t ws_size,
                              hipStream_t stream) {
  (void)n_in; (void)out_size; (void)d_ws; (void)ws_size;
  const float* logits = (const float*)d_in[0];
  const float* temps  = (const float*)d_in[1];
  const float* topps  = (const float*)d_in[2];
  const float* noise  = (const float*)d_in[3];
  int* out = (int*)d_out;

  const int rows = in_sizes[1];            // B = 256
  hipLaunchKernelGGL(topp_sample_kernel, dim3(rows), dim3(THREADS), 0, stream,
                     logits, temps, topps, noise, out);
}